// LSTMClassifier_41669772706604
// MI455X (gfx1250) — compile-verified
//
#include <hip/hip_runtime.h>
#include <hip/hip_bf16.h>

#define VOCAB 30522
#define EMB   768
#define HID   1024
#define G4    4096   // 4*HID
#define SEQ   4096
#define LABELS 2

#define KTILE 32
#define MTILE 128    // rows per block  (8 waves * 16)
#define NTILE 64     // cols per block  (4 subtiles * 16)
#define SCAN_BLOCKS 64   // each owns 16 hidden units

typedef __attribute__((ext_vector_type(2))) float v2f;
typedef __attribute__((ext_vector_type(8))) float v8f;

// ---------------------------------------------------------------------------
// Init: zero h double-buffer and the grid-barrier counter (re-run every call)
// ---------------------------------------------------------------------------
__global__ void lstm_init_kernel(float* __restrict__ hbuf, unsigned* __restrict__ bar) {
    int t = threadIdx.x;
    for (int i = t; i < 2 * HID; i += blockDim.x) hbuf[i] = 0.0f;
    if (t == 0) *bar = 0u;
}

// ---------------------------------------------------------------------------
// Phase 1: xg[T,4H] = gather(emb, sentence) @ w_ih.T + (b_ih + b_hh)
// fp32 WMMA 16x16x4. Block = 8 waves; wave -> 16x64 strip (4 accumulators).
// ---------------------------------------------------------------------------
__global__ __launch_bounds__(256) void xg_gemm_wmma(
    const int*   __restrict__ sentence,
    const float* __restrict__ emb,
    const float* __restrict__ w_ih,
    const float* __restrict__ b_ih,
    const float* __restrict__ b_hh,
    float*       __restrict__ xg)
{
    __shared__ int   tok[MTILE];
    __shared__ float xa[MTILE][KTILE + 2];   // stride 34: banks spread for frag reads
    __shared__ float wb[KTILE][NTILE + 1];   // stride 65

    const int tid   = threadIdx.x;
    const int lane  = tid & 31;
    const int wid   = tid >> 5;
    const int mTile = blockIdx.y * MTILE;
    const int nTile = blockIdx.x * NTILE;

    if (tid < MTILE) tok[tid] = sentence[mTile + tid];
    __syncthreads();

    v8f acc0 = {}, acc1 = {}, acc2 = {}, acc3 = {};

    const int rBase   = wid * 16;
    const int halfSel = lane >> 4;   // 0: lanes 0-15, 1: lanes 16-31
    const int l15     = lane & 15;

    for (int k0 = 0; k0 < EMB; k0 += KTILE) {
        // A tile: gathered embedding rows (row-major, coalesced 16-float runs)
        {
            const int r  = tid >> 1;
            const int kb = (tid & 1) * 16;
            const float* src = emb + (size_t)tok[r] * EMB + k0 + kb;
            #pragma unroll
            for (int i = 0; i < 16; ++i) xa[r][kb + i] = src[i];
        }
        // B tile transposed into LDS: wb[k][n] = w_ih[(nTile+n)*EMB + k0 + k]
        {
            const int n  = tid >> 2;
            const int kb = (tid & 3) * 8;
            const float* src = w_ih + (size_t)(nTile + n) * EMB + k0 + kb;
            #pragma unroll
            for (int i = 0; i < 8; ++i) wb[kb + i][n] = src[i];
        }
        __syncthreads();

        #pragma unroll
        for (int kk = 0; kk < KTILE; kk += 4) {
            const int kb = kk + (halfSel << 1);   // lanes 16-31 carry K+2 per ISA layout
            v2f a;  a.x  = xa[rBase + l15][kb];   a.y  = xa[rBase + l15][kb + 1];
            v2f b0; b0.x = wb[kb][ 0 + l15];      b0.y = wb[kb + 1][ 0 + l15];
            v2f b1; b1.x = wb[kb][16 + l15];      b1.y = wb[kb + 1][16 + l15];
            v2f b2; b2.x = wb[kb][32 + l15];      b2.y = wb[kb + 1][32 + l15];
            v2f b3; b3.x = wb[kb][48 + l15];      b3.y = wb[kb + 1][48 + l15];
            acc0 = __builtin_amdgcn_wmma_f32_16x16x4_f32(false, a, false, b0, (short)0, acc0, false, false);
            acc1 = __builtin_amdgcn_wmma_f32_16x16x4_f32(false, a, false, b1, (short)0, acc1, false, false);
            acc2 = __builtin_amdgcn_wmma_f32_16x16x4_f32(false, a, false, b2, (short)0, acc2, false, false);
            acc3 = __builtin_amdgcn_wmma_f32_16x16x4_f32(false, a, false, b3, (short)0, acc3, false, false);
        }
        __syncthreads();
    }

    // Epilogue: C layout -> VGPR r holds M=r (lanes 0-15) / M=r+8 (lanes 16-31)
    const int rowBase = mTile + rBase + halfSel * 8;
    #pragma unroll
    for (int r = 0; r < 8; ++r) {
        const size_t row = (size_t)(rowBase + r) * G4;
        int c0 = nTile +  0 + l15;
        int c1 = nTile + 16 + l15;
        int c2 = nTile + 32 + l15;
        int c3 = nTile + 48 + l15;
        xg[row + c0] = acc0[r] + b_ih[c0] + b_hh[c0];
        xg[row + c1] = acc1[r] + b_ih[c1] + b_hh[c1];
        xg[row + c2] = acc2[r] + b_ih[c2] + b_hh[c2];
        xg[row + c3] = acc3[r] + b_ih[c3] + b_hh[c3];
    }
}

// ---------------------------------------------------------------------------
// Grid-wide barrier: monotonic counter, step s waits for (s+1)*SCAN_BLOCKS.
// ---------------------------------------------------------------------------
__device__ __forceinline__ void grid_sync(unsigned* bar, unsigned target) {
    __threadfence();                 // release my stores to agent scope
    __syncthreads();
    if (threadIdx.x == 0) {
        __hip_atomic_fetch_add(bar, 1u, __ATOMIC_RELEASE, __HIP_MEMORY_SCOPE_AGENT);
        while (__hip_atomic_load(bar, __ATOMIC_ACQUIRE, __HIP_MEMORY_SCOPE_AGENT) < target)
            __builtin_amdgcn_s_sleep(1);
    }
    __syncthreads();
}

__device__ __forceinline__ float sigmoidf(float x) {
    return 1.0f / (1.0f + __expf(-x));
}

// ---------------------------------------------------------------------------
// Phase 2: persistent LSTM scan. Block b owns hidden units [16b, 16b+16):
// 64 rows of w_hh (i,f,g,o x 16), hot in L2. 4 lanes per row, 256 floats each.
// ---------------------------------------------------------------------------
__global__ __launch_bounds__(256) void lstm_scan_kernel(
    const float* __restrict__ w_hh,
    const float* __restrict__ xg,
    float*       __restrict__ hbuf,   // [2][HID] double buffer, buf0 zeroed
    unsigned*    __restrict__ bar)
{
    __shared__ float h_s[HID];
    __shared__ float gates_s[64];     // [gate*16 + j]
    __shared__ float c_s[16];

    const int tid     = threadIdx.x;
    const int row_idx = tid >> 2;          // 0..63  == gate*16 + j
    const int part    = tid & 3;           // 256-wide K slice
    const int gate    = row_idx >> 4;
    const int j       = row_idx & 15;
    const int u0      = blockIdx.x * 16;
    const int row     = gate * HID + u0 + j;   // row of w_hh / column of xg

    if (tid < 16) c_s[tid] = 0.0f;

    const float4* w4 = (const float4*)(w_hh + (size_t)row * HID + part * 256);

    for (int t = 0; t < SEQ; ++t) {
        const float* hOld = hbuf + (t & 1) * HID;
        float*       hNew = hbuf + ((t + 1) & 1) * HID;

        for (int i = tid; i < HID; i += 256) h_s[i] = hOld[i];
        __syncthreads();

        // partial dot over 256 elements (64 x float4 FMAs from LDS + L2)
        const float4* h4 = (const float4*)(h_s + part * 256);
        float p = 0.0f;
        #pragma unroll 8
        for (int i = 0; i < 64; ++i) {
            float4 a = w4[i], b = h4[i];
            p += a.x * b.x + a.y * b.y + a.z * b.z + a.w * b.w;
        }
        // reduce the 4-lane group (wave32 shuffles)
        p += __shfl_xor(p, 1, 32);
        p += __shfl_xor(p, 2, 32);

        if (part == 0)
            gates_s[row_idx] = xg[(size_t)t * G4 + row] + p;   // xg already has biases

        // prefetch next timestep's pre-activation (global_prefetch_b8)
        if (part == 0 && t + 1 < SEQ)
            __builtin_prefetch(&xg[(size_t)(t + 1) * G4 + row], 0, 0);

        __syncthreads();

        if (tid < 16) {
            const int jj = tid;
            float iv = sigmoidf(gates_s[     jj]);
            float fv = sigmoidf(gates_s[16 + jj]);
            float gv = tanhf   (gates_s[32 + jj]);
            float ov = sigmoidf(gates_s[48 + jj]);
            float c  = fv * c_s[jj] + iv * gv;
            c_s[jj]  = c;
            hNew[u0 + jj] = ov * tanhf(c);
        }

        grid_sync(bar, (unsigned)(t + 1) * SCAN_BLOCKS);
    }
}

// ---------------------------------------------------------------------------
// Phase 3: y = h_T @ w_out.T + b_out ; log_softmax -> out[2]
// ---------------------------------------------------------------------------
__global__ __launch_bounds__(256) void head_kernel(
    const float* __restrict__ h,      // final h (buffer 0)
    const float* __restrict__ w_out,
    const float* __restrict__ b_out,
    float*       __restrict__ out)
{
    __shared__ float red[256];
    const int t     = threadIdx.x;
    const int label = t >> 7;         // 0 or 1
    const int k     = t & 127;

    float p = 0.0f;
    for (int i = 0; i < HID; i += 128)
        p += h[k + i] * w_out[label * HID + k + i];
    red[t] = p;
    __syncthreads();
    for (int s = 64; s > 0; s >>= 1) {
        if (k < s) red[t] += red[t + s];
        __syncthreads();
    }
    if (t == 0) {
        float y0 = red[0]   + b_out[0];
        float y1 = red[128] + b_out[1];
        float m   = fmaxf(y0, y1);
        float lse = m + __logf(__expf(y0 - m) + __expf(y1 - m));
        out[0] = y0 - lse;
        out[1] = y1 - lse;
    }
}

// ---------------------------------------------------------------------------
extern "C" void kernel_launch(void* const* d_in, const int* in_sizes, int n_in,
                              void* d_out, int out_size, void* d_ws, size_t ws_size,
                              hipStream_t stream) {
    const int*   sentence = (const int*)  d_in[0];
    const float* emb      = (const float*)d_in[1];
    const float* w_ih     = (const float*)d_in[2];
    const float* w_hh     = (const float*)d_in[3];
    const float* b_ih     = (const float*)d_in[4];
    const float* b_hh     = (const float*)d_in[5];
    const float* w_out    = (const float*)d_in[6];
    const float* b_out    = (const float*)d_in[7];

    float*    xg   = (float*)d_ws;                       // SEQ*G4 = 64 MB
    float*    hbuf = xg + (size_t)SEQ * G4;              // 2*HID
    unsigned* bar  = (unsigned*)(hbuf + 2 * HID);        // 1

    lstm_init_kernel<<<1, 256, 0, stream>>>(hbuf, bar);

    xg_gemm_wmma<<<dim3(G4 / NTILE, SEQ / MTILE), 256, 0, stream>>>(
        sentence, emb, w_ih, b_ih, b_hh, xg);

    lstm_scan_kernel<<<SCAN_BLOCKS, 256, 0, stream>>>(w_hh, xg, hbuf, bar);

    // final h lands in buffer 0 (last write at t=4095 -> (t+1)&1 == 0)
    head_kernel<<<1, 256, 0, stream>>>(hbuf, w_out, b_out, (float*)d_out);
}